// GraphSAGE_86698209837070
// MI455X (gfx1250) — compile-verified
//
#include <hip/hip_runtime.h>

typedef __attribute__((ext_vector_type(2))) float v2f;
typedef __attribute__((ext_vector_type(4))) float v4f;
typedef __attribute__((ext_vector_type(8))) float v8f;

#define D 64
#define LDSW 68   // padded LDS row stride (floats): stride mod 64 banks = 4 -> conflict-free b64 frag reads

// ---------------------------------------------------------------- zero fill
__global__ void __launch_bounds__(256) zero_f32(float* __restrict__ p, long long n) {
  long long i = (long long)blockIdx.x * blockDim.x + threadIdx.x;
  long long stride = (long long)gridDim.x * blockDim.x;
  for (; i < n; i += stride) p[i] = 0.0f;
}

// ---------------------------------------------------------------- scatter-sum over edges
// One wave (32 lanes) per edge; lane l owns feature columns 2l, 2l+1.
// Wave read of x[src] = 32 lanes x 8B = 256B fully coalesced; adds land in L2 atomics.
__global__ void __launch_bounds__(256) sage_scatter(const float* __restrict__ X,
                                                    const long long* __restrict__ ei,
                                                    float* __restrict__ msum,
                                                    float* __restrict__ deg,
                                                    int E, int accum_deg) {
  int gw   = (int)(((long long)blockIdx.x * blockDim.x + threadIdx.x) >> 5);
  int lane = threadIdx.x & 31;
  if (gw >= E) return;
  int src = (int)ei[gw];                 // edge_index row 0
  int dst = (int)ei[(long long)E + gw];  // edge_index row 1
  v2f v = *(const v2f*)(X + (long long)src * D + lane * 2);
  float* base = msum + (long long)dst * D + lane * 2;
  atomicAdd(base,     v.x);
  atomicAdd(base + 1, v.y);
  if (accum_deg && lane == 0) atomicAdd(deg + dst, 1.0f);
}

// ---------------------------------------------------------------- fused SAGE GEMM
// Out[m, :] = act( (msum[m,:]/max(deg[m],1)) @ Wl  +  Xr[m,:] @ Wr  + bias )
// 256 threads = 8 waves; each wave computes a 16-row x 64-col strip with
// V_WMMA_F32_16X16X4_F32 (full fp32 precision, K stepped by 4).
// K loop unrolled by 4 (not fully): keeps in-flight A fragments ~16 VGPRs so the
// wave stays under 256 VGPRs (no s_set_vgpr_msb, 4 waves/SIMD occupancy).
__global__ void __launch_bounds__(256, 2) sage_gemm(const float* __restrict__ Msum,
                                                    const float* __restrict__ Deg,
                                                    const float* __restrict__ Xr,
                                                    const float* __restrict__ Wl,
                                                    const float* __restrict__ Wr,
                                                    const float* __restrict__ bias,
                                                    float* __restrict__ Out,
                                                    int n, int relu) {
  __shared__ float lwl[D * LDSW];  // transposed: lwl[col][k] = Wl[k][col]
  __shared__ float lwr[D * LDSW];

  int t = threadIdx.x;
#pragma unroll
  for (int i = 0; i < (D * D) / 256; ++i) {
    int idx = t + i * 256;
    int k = idx >> 6, c = idx & 63;
    lwl[c * LDSW + k] = Wl[idx];
    lwr[c * LDSW + k] = Wr[idx];
  }
  __syncthreads();

  int lane = t & 31;
  int wave = t >> 5;
  int row0 = (blockIdx.x * 8 + wave) * 16;

  int m  = lane & 15;          // A-row / B-col within tile (ISA 16x4 f32 layout)
  int kb = (lane >> 4) * 2;    // lanes 0-15 -> K+{0,1}; lanes 16-31 -> K+{2,3}

  long long row = row0 + m;
  if (row >= n) row = n - 1;   // clamp: WMMA needs EXEC all-1s; stores are guarded
  float rs = 1.0f / fmaxf(Deg[row], 1.0f);

  const float* mrow = Msum + row * D + kb;
  const float* xrow = Xr   + row * D + kb;

  v8f c0 = {0.f,0.f,0.f,0.f,0.f,0.f,0.f,0.f};
  v8f c1 = {0.f,0.f,0.f,0.f,0.f,0.f,0.f,0.f};
  v8f c2 = {0.f,0.f,0.f,0.f,0.f,0.f,0.f,0.f};
  v8f c3 = {0.f,0.f,0.f,0.f,0.f,0.f,0.f,0.f};

#pragma unroll 4
  for (int k0 = 0; k0 < D; k0 += 4) {
    v2f am = *(const v2f*)(mrow + k0);
    am = am * rs;                          // fused mean-divide
    v2f ax = *(const v2f*)(xrow + k0);

#define SAGE_TILE(ACC, N0)                                                          \
    {                                                                               \
      v2f bl = *(const v2f*)&lwl[((N0) + m) * LDSW + k0 + kb];                      \
      v2f br = *(const v2f*)&lwr[((N0) + m) * LDSW + k0 + kb];                      \
      ACC = __builtin_amdgcn_wmma_f32_16x16x4_f32(false, am, false, bl,             \
                                                  (short)0, ACC, false, false);     \
      ACC = __builtin_amdgcn_wmma_f32_16x16x4_f32(false, ax, false, br,             \
                                                  (short)0, ACC, false, false);     \
    }
    SAGE_TILE(c0, 0)
    SAGE_TILE(c1, 16)
    SAGE_TILE(c2, 32)
    SAGE_TILE(c3, 48)
#undef SAGE_TILE
  }

  // Epilogue: C/D layout -> VGPR j = row j (lanes 0-15) or row 8+j (lanes 16-31)
  int nc = lane & 15;
  int mb = row0 + (lane >> 4) * 8;
#define SAGE_EPI(ACC, N0)                                                           \
  {                                                                                 \
    float bv = bias[(N0) + nc];                                                     \
    _Pragma("unroll")                                                               \
    for (int j = 0; j < 8; ++j) {                                                   \
      float v = ACC[j] + bv;                                                        \
      if (relu) v = fmaxf(v, 0.0f);                                                 \
      int r = mb + j;                                                               \
      if (r < n) Out[(long long)r * D + (N0) + nc] = v;                             \
    }                                                                               \
  }
  SAGE_EPI(c0, 0)
  SAGE_EPI(c1, 16)
  SAGE_EPI(c2, 32)
  SAGE_EPI(c3, 48)
#undef SAGE_EPI
}

// ---------------------------------------------------------------- classifier: [N,64] @ [64,2] + bc
__global__ void __launch_bounds__(256) sage_classify(const float* __restrict__ H,
                                                     const float* __restrict__ Wc,
                                                     const float* __restrict__ bc,
                                                     float* __restrict__ out, int n) {
  int i = blockIdx.x * blockDim.x + threadIdx.x;
  if (i >= n) return;
  const float* h = H + (long long)i * D;
  float a = bc[0], b = bc[1];
#pragma unroll
  for (int k = 0; k < D; k += 4) {
    v4f v = *(const v4f*)(h + k);
    a += v.x * Wc[2*k]   + v.y * Wc[2*k+2] + v.z * Wc[2*k+4] + v.w * Wc[2*k+6];
    b += v.x * Wc[2*k+1] + v.y * Wc[2*k+3] + v.z * Wc[2*k+5] + v.w * Wc[2*k+7];
  }
  out[2 * (long long)i]     = a;
  out[2 * (long long)i + 1] = b;
}

// ---------------------------------------------------------------- launch
extern "C" void kernel_launch(void* const* d_in, const int* in_sizes, int n_in,
                              void* d_out, int out_size, void* d_ws, size_t ws_size,
                              hipStream_t stream) {
  const float*     x   = (const float*)d_in[0];
  const long long* ei  = (const long long*)d_in[1];
  const float*     W1l = (const float*)d_in[2];
  const float*     b1  = (const float*)d_in[3];
  const float*     W1r = (const float*)d_in[4];
  const float*     W2l = (const float*)d_in[5];
  const float*     b2  = (const float*)d_in[6];
  const float*     W2r = (const float*)d_in[7];
  const float*     Wc  = (const float*)d_in[8];
  const float*     bc  = (const float*)d_in[9];

  const int N = in_sizes[0] / D;     // 100000
  const int E = in_sizes[1] / 2;     // 1.6M edges

  float* out  = (float*)d_out;
  float* msum = (float*)d_ws;                    // N*64  (reused by both layers)
  float* deg  = msum + (size_t)N * D;            // N     (contiguous with msum for one zero pass)
  float* h1   = deg + N;                         // N*64
  float* h2   = h1 + (size_t)N * D;              // N*64   => ~77 MB total workspace

  const int gemm_blocks    = (N + 127) / 128;    // 8 waves x 16 rows per block
  const int scatter_blocks = (E + 7) / 8;        // one wave per edge, 8 waves/block
  const int cls_blocks     = (N + 255) / 256;

  // ---- layer 1: h1 = relu(mean_agg(x) @ W1l + b1 + x @ W1r)
  zero_f32<<<2048, 256, 0, stream>>>(msum, (long long)N * D + N);   // msum + deg
  sage_scatter<<<scatter_blocks, 256, 0, stream>>>(x, ei, msum, deg, E, 1);
  sage_gemm<<<gemm_blocks, 256, 0, stream>>>(msum, deg, x, W1l, W1r, b1, h1, N, 1);

  // ---- layer 2: h2 = mean_agg(h1) @ W2l + b2 + h1 @ W2r   (deg reused)
  zero_f32<<<2048, 256, 0, stream>>>(msum, (long long)N * D);
  sage_scatter<<<scatter_blocks, 256, 0, stream>>>(h1, ei, msum, deg, E, 0);
  sage_gemm<<<gemm_blocks, 256, 0, stream>>>(msum, deg, h1, W2l, W2r, b2, h2, N, 0);

  // ---- classifier: out = h2 @ Wc + bc
  sage_classify<<<cls_blocks, 256, 0, stream>>>(h2, Wc, bc, out, N);
}